// ResidualGNNLayer_33071248179792
// MI455X (gfx1250) — compile-verified
//
#include <hip/hip_runtime.h>
#include <hip/hip_bf16.h>

#define D_DIM 256
#define LN_EPS 1e-3f

typedef __attribute__((ext_vector_type(2))) float v2f;
typedef __attribute__((ext_vector_type(8))) float v8f;

// ---------------- degree: deg[dst[e]] += 1 ----------------
__global__ void deg_kernel(const int* __restrict__ dst, float* __restrict__ deg, int E) {
    int e = blockIdx.x * blockDim.x + threadIdx.x;
    if (e < E) atomicAdd(&deg[dst[e]], 1.0f);
}

// ---------------- norm = rsqrt(deg + 1)  (self-loop) ----------------
__global__ void norm_kernel(float* __restrict__ deg, int n) {
    int i = blockIdx.x * blockDim.x + threadIdx.x;
    if (i < n) deg[i] = rsqrtf(deg[i] + 1.0f);
}

// ---------------- h = x @ W via V_WMMA_F32_16X16X4_F32 ----------------
// One wave (32 threads) computes a 16x16 tile of h. K loop in steps of 4.
// A-frag (16x4 f32): lane l -> row m = l&15, v0=A[m][2*(l>>4)], v1=A[m][2*(l>>4)+1]
// B-frag (4x16 f32): lane l -> col n = l&15, v0=B[2*(l>>4)][n], v1=B[2*(l>>4)+1][n]
// C/D (16x16 f32):   lane l, vgpr j -> M = j + 8*(l>>4), N = l&15
__global__ void gemm_wmma_kernel(const float* __restrict__ x,
                                 const float* __restrict__ W,
                                 float* __restrict__ h) {
    const int d = D_DIM;
    const int colTiles = d / 16;               // 16
    int tile = blockIdx.x;
    int colT = tile % colTiles;
    int rowT = tile / colTiles;
    int lane = threadIdx.x;                    // 0..31, full wave, EXEC all 1s
    int m    = lane & 15;
    int half = lane >> 4;                      // 0 or 1

    const float* arow = x + (size_t)(rowT * 16 + m) * d;   // A row for this lane
    int ncol = colT * 16 + m;                              // B/D column for this lane

    v8f acc = {};
    for (int kk = 0; kk < d; kk += 4) {
        int ka = kk + 2 * half;
        v2f a = *(const v2f*)(arow + ka);                  // 8B-aligned float2
        v2f b;
        b.x = W[(size_t)ka * d + ncol];
        b.y = W[(size_t)(ka + 1) * d + ncol];
        acc = __builtin_amdgcn_wmma_f32_16x16x4_f32(
            /*neg_a=*/false, a, /*neg_b=*/false, b,
            /*c_mod=*/(short)0, acc, /*reuse_a=*/false, /*reuse_b=*/false);
    }

    float* orow = h + (size_t)(rowT * 16) * d + ncol;
#pragma unroll
    for (int j = 0; j < 8; ++j) {
        orow[(size_t)(j + 8 * half) * d] = acc[j];
    }
}

// ---------------- scatter: agg[dst] += h[src] * norm[src] ----------------
// 64 threads per edge, float4 chunk per thread.
__global__ void scatter_kernel(const float* __restrict__ h,
                               const int* __restrict__ src,
                               const int* __restrict__ dst,
                               const float* __restrict__ norm,
                               float* __restrict__ agg, int E) {
    int t = blockIdx.x * blockDim.x + threadIdx.x;
    int e = t >> 6;
    if (e >= E) return;
    int c = (t & 63) << 2;                     // float offset within row
    int s  = src[e];
    int dd = dst[e];
    float ns = norm[s];
    float4 v = *(const float4*)(h + (size_t)s * D_DIM + c);
    float* op = agg + (size_t)dd * D_DIM + c;
    atomicAdd(op + 0, v.x * ns);
    atomicAdd(op + 1, v.y * ns);
    atomicAdd(op + 2, v.z * ns);
    atomicAdd(op + 3, v.w * ns);
}

// ---------------- finalize: residual + bias + self-loop + LayerNorm ----------------
// One 256-thread block per row; agg lives in `out` and is overwritten in place.
__global__ void finalize_kernel(const float* __restrict__ x,
                                const float* __restrict__ h,
                                const float* __restrict__ bias,
                                const float* __restrict__ gamma,
                                const float* __restrict__ beta,
                                const float* __restrict__ norm,
                                float* __restrict__ out) {
    int row = blockIdx.x;
    int c   = threadIdx.x;                     // 0..255
    size_t idx = (size_t)row * D_DIM + c;
    float ni = norm[row];
    float y = x[idx] + out[idx] * ni + h[idx] * (ni * ni) + bias[c];

    __shared__ float red[8];
    // block-wide sum for mean (wave32 shuffle + LDS across 8 waves)
    float s = y;
#pragma unroll
    for (int off = 16; off > 0; off >>= 1) s += __shfl_xor(s, off, 32);
    if ((threadIdx.x & 31) == 0) red[threadIdx.x >> 5] = s;
    __syncthreads();
    float total = 0.0f;
#pragma unroll
    for (int i = 0; i < 8; ++i) total += red[i];
    float mean = total * (1.0f / D_DIM);
    float dv = y - mean;
    __syncthreads();                           // protect red[] reuse

    float s2 = dv * dv;
#pragma unroll
    for (int off = 16; off > 0; off >>= 1) s2 += __shfl_xor(s2, off, 32);
    if ((threadIdx.x & 31) == 0) red[threadIdx.x >> 5] = s2;
    __syncthreads();
    float tot2 = 0.0f;
#pragma unroll
    for (int i = 0; i < 8; ++i) tot2 += red[i];
    float var = tot2 * (1.0f / D_DIM);

    out[idx] = gamma[c] * dv * rsqrtf(var + LN_EPS) + beta[c];
}

extern "C" void kernel_launch(void* const* d_in, const int* in_sizes, int n_in,
                              void* d_out, int out_size, void* d_ws, size_t ws_size,
                              hipStream_t stream) {
    const float* x     = (const float*)d_in[0];
    const int*   edge  = (const int*)  d_in[1];
    const float* W     = (const float*)d_in[2];
    const float* bias  = (const float*)d_in[3];
    const float* gamma = (const float*)d_in[4];
    const float* beta  = (const float*)d_in[5];
    float* out = (float*)d_out;

    int n = in_sizes[0] / D_DIM;               // 100000
    int E = in_sizes[1] / 2;                   // 1600000
    const int* src = edge;
    const int* dst = edge + E;

    float* h    = (float*)d_ws;                // N*D floats
    float* norm = h + (size_t)n * D_DIM;       // N floats (deg -> norm in place)

    // zero the agg accumulator (d_out) and degree buffer
    hipMemsetAsync(out,  0, (size_t)n * D_DIM * sizeof(float), stream);
    hipMemsetAsync(norm, 0, (size_t)n * sizeof(float), stream);

    deg_kernel <<<(E + 255) / 256, 256, 0, stream>>>(dst, norm, E);
    norm_kernel<<<(n + 255) / 256, 256, 0, stream>>>(norm, n);

    // one 32-thread wave per 16x16 tile: (N/16) * (D/16) blocks
    gemm_wmma_kernel<<<(n / 16) * (D_DIM / 16), 32, 0, stream>>>(x, W, h);

    int totalT = E * 64;                       // 102.4M threads, fits in int
    scatter_kernel<<<(totalT + 255) / 256, 256, 0, stream>>>(h, src, dst, norm, out, E);

    finalize_kernel<<<n, D_DIM, 0, stream>>>(x, h, bias, gamma, beta, norm, out);
}